// TransformerBlock_90348932039017
// MI455X (gfx1250) — compile-verified
//
#include <hip/hip_runtime.h>
#include <math.h>

typedef __attribute__((ext_vector_type(2))) float v2f;
typedef __attribute__((ext_vector_type(8))) float v8f;

static __device__ __forceinline__ float gelu_exact(float x) {
  return 0.5f * x * (1.0f + erff(x * 0.70710678118654752f));
}
static __device__ __forceinline__ float sigmoidf_(float x) {
  return 1.0f / (1.0f + __expf(-x));
}

// ---------------- LayerNorm over channel dim ----------------
__global__ void k_ln(const float* __restrict__ in, const float* __restrict__ g,
                     const float* __restrict__ be, float* __restrict__ out,
                     int total /*B*HW*/, int C, int HW) {
  int idx = blockIdx.x * blockDim.x + threadIdx.x;
  if (idx >= total) return;
  int b = idx / HW, p = idx % HW;
  const float* base = in + (size_t)b * C * HW + p;
  float s = 0.f, ss = 0.f;
  for (int c = 0; c < C; ++c) { float v = base[(size_t)c * HW]; s += v; ss += v * v; }
  float mu = s / C;
  float var = ss / C - mu * mu;
  float inv = rsqrtf(var + 1e-5f);
  float* ob = out + (size_t)b * C * HW + p;
  for (int c = 0; c < C; ++c) {
    float v = base[(size_t)c * HW];
    ob[(size_t)c * HW] = (v - mu) * inv * g[c] + be[c];
  }
}

// ---------------- Direct 3x3 conv (pad=1), optional exact GELU ----------------
__global__ void k_conv3x3(const float* __restrict__ in, const float* __restrict__ w,
                          const float* __restrict__ bias, float* __restrict__ out,
                          int B, int Cin, int Cout, int H, int W, int act) {
  int idx = blockIdx.x * blockDim.x + threadIdx.x;
  int total = B * Cout * H * W;
  if (idx >= total) return;
  int x = idx % W, y = (idx / W) % H, oc = (idx / (W * H)) % Cout, b = idx / (W * H * Cout);
  float acc = bias[oc];
  const float* inb = in + (size_t)b * Cin * H * W;
  for (int ci = 0; ci < Cin; ++ci) {
    const float* ip = inb + (size_t)ci * H * W;
    const float* wp = w + ((size_t)oc * Cin + ci) * 9;
    for (int ky = 0; ky < 3; ++ky) {
      int iy = y + ky - 1; if ((unsigned)iy >= (unsigned)H) continue;
      for (int kx = 0; kx < 3; ++kx) {
        int ix = x + kx - 1; if ((unsigned)ix >= (unsigned)W) continue;
        acc += wp[ky * 3 + kx] * ip[iy * W + ix];
      }
    }
  }
  out[idx] = act ? gelu_exact(acc) : acc;
}

// ---------------- SE: per-(b,c) global mean pooling ----------------
__global__ void k_se_pool(const float* __restrict__ c2, float* __restrict__ mean, int HW) {
  __shared__ float sh[256];
  int bc = blockIdx.x;
  const float* p = c2 + (size_t)bc * HW;
  float s = 0.f;
  for (int i = threadIdx.x; i < HW; i += 256) s += p[i];
  sh[threadIdx.x] = s; __syncthreads();
  for (int st = 128; st > 0; st >>= 1) {
    if ((int)threadIdx.x < st) sh[threadIdx.x] += sh[threadIdx.x + st];
    __syncthreads();
  }
  if (threadIdx.x == 0) mean[bc] = sh[0] / HW;
}

// SE gate: 32->1 relu, 1->32 sigmoid (EMBED//18 == 1). One wave32 per batch.
__global__ void k_se_gate(const float* __restrict__ mean, const float* __restrict__ aw1,
                          const float* __restrict__ ab1, const float* __restrict__ aw2,
                          const float* __restrict__ ab2, float* __restrict__ sey) {
  int b = blockIdx.x, c = threadIdx.x;
  float v = mean[b * 32 + c] * aw1[c];
  v += __shfl_xor(v, 1); v += __shfl_xor(v, 2); v += __shfl_xor(v, 4);
  v += __shfl_xor(v, 8); v += __shfl_xor(v, 16);
  float y1 = fmaxf(v + ab1[0], 0.f);
  sey[b * 32 + c] = sigmoidf_(aw2[c] * y1 + ab2[c]);
}

// ---------------- 1x1 conv as WMMA f32 16x16x4 GEMM (compile-time M,K) ----------
// out[b,m,n] = sum_k W[m,k]*in[b,k,n] + bias[m] (+ res[b,m,n])
// One wave handles a 16-wide column tile for ALL row tiles, sharing B loads.
template <int M, int K>
__global__ void __launch_bounds__(32)
k_gemm1x1(const float* __restrict__ W, const float* __restrict__ bias,
          const float* __restrict__ in, const float* __restrict__ res,
          float* __restrict__ out, int N) {
  constexpr int RT = (M + 15) / 16;   // row tiles
  constexpr int KC = K / 4;           // K chunks of 4
  int colT = blockIdx.x;
  int b = blockIdx.y;
  const float* inb = in + (size_t)b * K * N;
  float* outb = out + (size_t)b * M * N;
  int l = threadIdx.x, half = l >> 4, lane = l & 15;
  int col = colT * 16 + lane;

  // Preload all A (weight) operands: branchless clamped index + float mask.
  v2f Aop[RT][KC];
#pragma unroll
  for (int rt = 0; rt < RT; ++rt) {
    int ra = rt * 16 + lane;
    float msk = (ra < M) ? 1.f : 0.f;
    int ri = (ra < M) ? ra : 0;
#pragma unroll
    for (int kc = 0; kc < KC; ++kc) {
      int ka = kc * 4 + 2 * half;     // half->K mapping per ISA A layout
      Aop[rt][kc].x = W[ri * K + ka] * msk;
      Aop[rt][kc].y = W[ri * K + ka + 1] * msk;
    }
  }

  v8f acc[RT];
#pragma unroll
  for (int rt = 0; rt < RT; ++rt)
#pragma unroll
    for (int e = 0; e < 8; ++e) acc[rt][e] = 0.f;

#pragma unroll
  for (int kc = 0; kc < KC; ++kc) {
    int ka = kc * 4 + 2 * half;       // same half->K mapping for B operand
    v2f bb;
    bb.x = inb[(size_t)ka * N + col];
    bb.y = inb[(size_t)(ka + 1) * N + col];
#pragma unroll
    for (int rt = 0; rt < RT; ++rt)
      acc[rt] = __builtin_amdgcn_wmma_f32_16x16x4_f32(false, Aop[rt][kc], false, bb,
                                                      (short)0, acc[rt], false, false);
  }

#pragma unroll
  for (int rt = 0; rt < RT; ++rt) {
#pragma unroll
    for (int v = 0; v < 8; ++v) {
      int row = rt * 16 + v + 8 * half;   // D layout: VGPR v = rows v / v+8
      if (row < M) {
        float r = acc[rt][v] + bias[row];
        if (res) r += res[(size_t)b * M * N + (size_t)row * N + col];
        outb[(size_t)row * N + col] = r;
      }
    }
  }
}

// ---------------- sq2: grouped 2x2 stride-2 conv (8ch 96x96 -> 16ch 48x48) ----------------
__global__ void k_sq2(const float* __restrict__ in, const float* __restrict__ w,
                      const float* __restrict__ bias, float* __restrict__ out) {
  int idx = blockIdx.x * blockDim.x + threadIdx.x;
  if (idx >= 2 * 16 * 48 * 48) return;
  int x = idx % 48, y = (idx / 48) % 48, oc = (idx / 2304) % 16, b = idx / (2304 * 16);
  const float* ip = in + ((size_t)b * 8 + (oc >> 1)) * 9216;
  const float* wp = w + oc * 4;
  int iy = y * 2, ix = x * 2;
  out[idx] = bias[oc] + wp[0] * ip[iy * 96 + ix] + wp[1] * ip[iy * 96 + ix + 1]
           + wp[2] * ip[(iy + 1) * 96 + ix] + wp[3] * ip[(iy + 1) * 96 + ix + 1];
}

// ---------------- channel attention (hd=2, e=2 softmax; IDX permute folded) ----------------
__global__ void k_chan_attn(const float* __restrict__ z, const float* __restrict__ qkvw,
                            const float* __restrict__ qkvb, const float* __restrict__ t,
                            float* __restrict__ o) {
  const int n = 2304;
  int h = blockIdx.x, b = blockIdx.y, tid = threadIdx.x;
  __shared__ float sh[256];
  __shared__ float aa[4];
  float qw[2], qb[2], kw[2], kb[2], vw[2], vb[2];
  const float* zc[2];
#pragma unroll
  for (int d = 0; d < 2; ++d) {
    int flat = 2 * h + d, e = flat >> 4, c = flat & 15;
    int cz = (c >> 1) + ((c & 1) << 3);           // IDX channel map
    zc[d] = z + ((size_t)b * 16 + cz) * n;
    qw[d] = qkvw[c * 6 + e];     qb[d] = qkvb[c * 6 + e];
    kw[d] = qkvw[c * 6 + 2 + e]; kb[d] = qkvb[c * 6 + 2 + e];
    vw[d] = qkvw[c * 6 + 4 + e]; vb[d] = qkvb[c * 6 + 4 + e];
  }
  float red[8];
#pragma unroll
  for (int i = 0; i < 8; ++i) red[i] = 0.f;
  for (int i = tid; i < n; i += 256) {
    float z0 = zc[0][i], z1 = zc[1][i];
    float q0 = z0 * qw[0] + qb[0], q1 = z1 * qw[1] + qb[1];
    float k0 = z0 * kw[0] + kb[0], k1 = z1 * kw[1] + kb[1];
    red[0] += q0 * k0; red[1] += q0 * k1; red[2] += q1 * k0; red[3] += q1 * k1;
    red[4] += q0 * q0; red[5] += q1 * q1; red[6] += k0 * k0; red[7] += k1 * k1;
  }
  float tot[8];
  for (int q = 0; q < 8; ++q) {
    sh[tid] = red[q]; __syncthreads();
    for (int st = 128; st > 0; st >>= 1) {
      if (tid < st) sh[tid] += sh[tid + st];
      __syncthreads();
    }
    tot[q] = sh[0]; __syncthreads();
  }
  if (tid == 0) {
    float th = t[h];
    float qn0 = fmaxf(sqrtf(tot[4]), 1e-12f), qn1 = fmaxf(sqrtf(tot[5]), 1e-12f);
    float kn0 = fmaxf(sqrtf(tot[6]), 1e-12f), kn1 = fmaxf(sqrtf(tot[7]), 1e-12f);
    float g00 = tot[0] / (qn0 * kn0) * th, g01 = tot[1] / (qn0 * kn1) * th;
    float g10 = tot[2] / (qn1 * kn0) * th, g11 = tot[3] / (qn1 * kn1) * th;
    float m0 = fmaxf(g00, g01), e0 = __expf(g00 - m0), e1 = __expf(g01 - m0);
    aa[0] = e0 / (e0 + e1); aa[1] = e1 / (e0 + e1);
    float m1 = fmaxf(g10, g11), f0 = __expf(g10 - m1), f1 = __expf(g11 - m1);
    aa[2] = f0 / (f0 + f1); aa[3] = f1 / (f0 + f1);
  }
  __syncthreads();
  float a00 = aa[0], a01 = aa[1], a10 = aa[2], a11 = aa[3];
  float* ob = o + ((size_t)b * 16 + h) * 2 * n;
  for (int i = tid; i < n; i += 256) {
    float z0 = zc[0][i], z1 = zc[1][i];
    float v0 = z0 * vw[0] + vb[0], v1 = z1 * vw[1] + vb[1];
    ob[i]     = a00 * v0 + a01 * v1;
    ob[n + i] = a10 * v0 + a11 * v1;
  }
}

// ---------------- fuse conv (groups=C, EXP=2): shared by ca/sa ----------------
__global__ void k_fuse(const float* __restrict__ oraw, const float* __restrict__ w,
                       const float* __restrict__ bias, float* __restrict__ out, int total) {
  int idx = blockIdx.x * blockDim.x + threadIdx.x;
  if (idx >= total) return;
  const int n = 2304;
  int i = idx % n, c = (idx / n) & 15, b = idx / (n * 16);
  const float* ob = oraw + (size_t)b * 16 * 2 * n;
  float o0 = ob[((size_t)((c >> 1) * 2 + (c & 1))) * n + i];          // e=0
  float o1 = ob[((size_t)((8 + (c >> 1)) * 2 + (c & 1))) * n + i];    // e=1
  out[idx] = w[c * 2 + 0] * o0 + w[c * 2 + 1] * o1 + bias[c];
}

// ---------------- spatial flash-attention with WMMA f32 16x16x4 ----------------
// one wave per (b, h, 16-query tile); NIDX permute + qkv depthwise folded in.
// Q,K are l2-normalized => scores = t*cos ∈ [-|t|,|t|]; softmax uses the STATIC
// shift M=|t| (shift-invariant, no overflow), so the inner loop has NO cross-lane
// reductions at all -- just WMMA + exp + per-lane FMA accumulation.
__global__ void __launch_bounds__(32)
k_spat_attn(const float* __restrict__ z, const float* __restrict__ qkvw,
            const float* __restrict__ qkvb, const float* __restrict__ t,
            float* __restrict__ o) {
  const int n = 2304;
  int qT = blockIdx.x, h = blockIdx.y, b = blockIdx.z;
  int l = threadIdx.x, half = l >> 4, lane = l & 15;

  float qw[2], qb[2], kw[2], kb[2], vw[2], vb[2];
  const float* zc[2];
#pragma unroll
  for (int d = 0; d < 2; ++d) {
    int flat = 2 * h + d, e = flat >> 4, c = flat & 15;
    int cz = (c >> 3) + ((c & 7) << 1);           // NIDX channel map
    zc[d] = z + ((size_t)b * 16 + cz) * n;
    qw[d] = qkvw[c * 6 + e];     qb[d] = qkvb[c * 6 + e];
    kw[d] = qkvw[c * 6 + 2 + e]; kb[d] = qkvb[c * 6 + 2 + e];
    vw[d] = qkvw[c * 6 + 4 + e]; vb[d] = qkvb[c * 6 + 4 + e];
  }
  float th = t[h];
  float M = fabsf(th);                // static softmax shift (|score| <= |t|)

  // A operand: normalized q rows (K=2,3 -> zero in upper half)
  v2f A = {};
  {
    int qrow = qT * 16 + lane;
    float q0 = zc[0][qrow] * qw[0] + qb[0];
    float q1 = zc[1][qrow] * qw[1] + qb[1];
    float inv = rsqrtf(fmaxf(q0 * q0 + q1 * q1, 1e-24f));  // 1/max(||q||,1e-12)
    float m = (half == 0) ? 1.f : 0.f;
    A.x = q0 * inv * m;
    A.y = q1 * inv * m;
  }

  float rs[8], a0[8], a1[8];
#pragma unroll
  for (int v = 0; v < 8; ++v) { rs[v] = 0.f; a0[v] = 0.f; a1[v] = 0.f; }

  for (int mb = 0; mb < n; mb += 16) {
    int col = mb + lane;
    float z0 = zc[0][col], z1 = zc[1][col];
    float k0 = z0 * kw[0] + kb[0];
    float k1 = z1 * kw[1] + kb[1];
    float inv = rsqrtf(fmaxf(k0 * k0 + k1 * k1, 1e-24f));
    float m = (half == 0) ? 1.f : 0.f;
    v2f Bv;
    Bv.x = k0 * inv * m;
    Bv.y = k1 * inv * m;
    float v0 = z0 * vw[0] + vb[0];
    float v1 = z1 * vw[1] + vb[1];
    v8f S;
#pragma unroll
    for (int e = 0; e < 8; ++e) S[e] = 0.f;
    S = __builtin_amdgcn_wmma_f32_16x16x4_f32(false, A, false, Bv, (short)0, S,
                                              false, false);
#pragma unroll
    for (int v = 0; v < 8; ++v) {
      float p = __expf(fmaf(S[v], th, -M));   // exp(score - M), <= 1
      rs[v] += p;
      a0[v] += p * v0;
      a1[v] += p * v1;
    }
  }
  // one-time epilogue reduction within each 16-lane half
  size_t ob = ((size_t)b * 16 + h) * 2 * (size_t)n;
#pragma unroll
  for (int v = 0; v < 8; ++v) {
    float s0 = a0[v], s1 = a1[v], ss = rs[v];
    s0 += __shfl_xor(s0, 1); s0 += __shfl_xor(s0, 2); s0 += __shfl_xor(s0, 4); s0 += __shfl_xor(s0, 8);
    s1 += __shfl_xor(s1, 1); s1 += __shfl_xor(s1, 2); s1 += __shfl_xor(s1, 4); s1 += __shfl_xor(s1, 8);
    ss += __shfl_xor(ss, 1); ss += __shfl_xor(ss, 2); ss += __shfl_xor(ss, 4); ss += __shfl_xor(ss, 8);
    if (lane == 0) {
      int row = qT * 16 + v + 8 * half;           // D layout: VGPR v = rows v / v+8
      float invs = 1.f / ss;
      o[ob + row]     = s0 * invs;
      o[ob + n + row] = s1 * invs;
    }
  }
}

// ---------------- un1: grouped 1x1 (16 -> 32, groups=8) ----------------
__global__ void k_un1(const float* __restrict__ z, const float* __restrict__ w,
                      const float* __restrict__ bias, float* __restrict__ out, int total) {
  int idx = blockIdx.x * blockDim.x + threadIdx.x;
  if (idx >= total) return;
  const int n = 2304;
  int i = idx % n, oc = (idx / n) & 31, b = idx / (n * 32);
  int g = oc >> 2;
  const float* zb = z + (size_t)b * 16 * n;
  out[idx] = w[oc * 2] * zb[(size_t)(g * 2) * n + i]
           + w[oc * 2 + 1] * zb[(size_t)(g * 2 + 1) * n + i] + bias[oc];
}

// ---------------- pixel shuffle r=2: (B,32,48,48) -> (B,8,96,96) ----------------
__global__ void k_shuffle(const float* __restrict__ in, float* __restrict__ out, int total) {
  int idx = blockIdx.x * blockDim.x + threadIdx.x;
  if (idx >= total) return;
  int x = idx % 96, y = (idx / 96) % 96, c = (idx / 9216) & 7, b = idx / (9216 * 8);
  int ic = c * 4 + (y & 1) * 2 + (x & 1);
  out[idx] = in[(((size_t)b * 32 + ic) * 48 + (y >> 1)) * 48 + (x >> 1)];
}

// ---------------- residual: x2 = x + z + c2*se_y ----------------
__global__ void k_resadd(const float* __restrict__ x, const float* __restrict__ z,
                         const float* __restrict__ c2, const float* __restrict__ sey,
                         float* __restrict__ out, int total) {
  int idx = blockIdx.x * blockDim.x + threadIdx.x;
  if (idx >= total) return;
  int c = (idx / 9216) & 31, b = idx / (9216 * 32);
  out[idx] = x[idx] + z[idx] + c2[idx] * sey[b * 32 + c];
}

// ---------------- fused depthwise 7x7 (32->2x32) + gelu*sigmoid gate ----------------
__global__ void k_dwgate(const float* __restrict__ in, const float* __restrict__ w,
                         const float* __restrict__ bias, float* __restrict__ out, int total) {
  int idx = blockIdx.x * blockDim.x + threadIdx.x;
  if (idx >= total) return;
  int x = idx % 96, y = (idx / 96) % 96, c = (idx / 9216) & 31, b = idx / (9216 * 32);
  const float* ip = in + ((size_t)b * 32 + c) * 9216;
  const float* w0 = w + (size_t)(2 * c) * 49;
  const float* w1 = w0 + 49;
  float acc0 = bias[2 * c], acc1 = bias[2 * c + 1];
  for (int ky = 0; ky < 7; ++ky) {
    int iy = y + ky - 3; if ((unsigned)iy >= 96u) continue;
    for (int kx = 0; kx < 7; ++kx) {
      int ix = x + kx - 3; if ((unsigned)ix >= 96u) continue;
      float v = ip[iy * 96 + ix];
      acc0 += w0[ky * 7 + kx] * v;
      acc1 += w1[ky * 7 + kx] * v;
    }
  }
  out[idx] = gelu_exact(acc0) * sigmoidf_(acc1);
}

// ================= host side =================
static inline int cdiv(int a, int b) { return (a + b - 1) / b; }

extern "C" void kernel_launch(void* const* d_in, const int* in_sizes, int n_in,
                              void* d_out, int out_size, void* d_ws, size_t ws_size,
                              hipStream_t stream) {
  const float* x       = (const float*)d_in[0];
  const float* cab_w1  = (const float*)d_in[1];
  const float* cab_w2  = (const float*)d_in[2];
  const float* cab_aw1 = (const float*)d_in[3];
  const float* cab_aw2 = (const float*)d_in[4];
  const float* sq1_w   = (const float*)d_in[5];
  const float* sq2_w   = (const float*)d_in[6];
  const float* ca_qkv_w= (const float*)d_in[7];
  const float* ca_fus_w= (const float*)d_in[8];
  const float* sa_qkv_w= (const float*)d_in[9];
  const float* sa_fus_w= (const float*)d_in[10];
  const float* un1_w   = (const float*)d_in[11];
  const float* un2_w   = (const float*)d_in[12];
  const float* da_w1   = (const float*)d_in[13];
  const float* da_dw_w = (const float*)d_in[14];
  const float* da_pw   = (const float*)d_in[15];
  const float* cab_b1  = (const float*)d_in[16];
  const float* cab_b2  = (const float*)d_in[17];
  const float* cab_ab1 = (const float*)d_in[18];
  const float* cab_ab2 = (const float*)d_in[19];
  const float* sq1_b   = (const float*)d_in[20];
  const float* sq2_b   = (const float*)d_in[21];
  const float* ca_qkv_b= (const float*)d_in[22];
  const float* ca_fus_b= (const float*)d_in[23];
  const float* sa_qkv_b= (const float*)d_in[24];
  const float* sa_fus_b= (const float*)d_in[25];
  const float* un1_b   = (const float*)d_in[26];
  const float* un2_b   = (const float*)d_in[27];
  const float* da_b1   = (const float*)d_in[28];
  const float* da_dw_b = (const float*)d_in[29];
  const float* da_pb   = (const float*)d_in[30];
  const float* ln1_w   = (const float*)d_in[31];
  const float* ln1_b   = (const float*)d_in[32];
  const float* ln2_w   = (const float*)d_in[33];
  const float* ln2_b   = (const float*)d_in[34];
  const float* ca_t    = (const float*)d_in[35];
  const float* sa_t    = (const float*)d_in[36];
  float* out = (float*)d_out;

  const int Bn = 2, HW = 9216, n2 = 2304;
  float* ws = (float*)d_ws;
  float* xn    = ws;                 // 589824
  float* c1    = ws + 589824;        // 147456
  float* c2    = ws + 737280;        // 589824
  float* sem   = ws + 1327104;       // 64
  float* sey   = ws + 1327168;       // 64
  float* zsq1  = ws + 1327232;       // 147456
  float* zsq2  = ws + 1474688;       // 73728
  float* oca   = ws + 1548416;       // 147456
  float* zca   = ws + 1695872;       // 73728
  float* osa   = ws + 1769600;       // 147456
  float* zsa   = ws + 1917056;       // 73728
  float* un1o  = ws + 1990784;       // 147456
  float* shuf  = ws + 2138240;       // 147456
  float* zfin  = ws + 2285696;       // 589824
  float* x2    = ws + 2875520;       // 589824
  float* xn2   = ws + 3465344;       // 589824
  float* d1    = ws + 4055168;       // 589824
  float* dact  = ws + 4644992;       // 589824  (end = 5234816 floats ~= 21 MB)

  // 1. LN1
  k_ln<<<cdiv(Bn * HW, 256), 256, 0, stream>>>(x, ln1_w, ln1_b, xn, Bn * HW, 32, HW);
  // 2-3. CAB convs
  k_conv3x3<<<cdiv(Bn * 8 * HW, 256), 256, 0, stream>>>(xn, cab_w1, cab_b1, c1, Bn, 32, 8, 96, 96, 1);
  k_conv3x3<<<cdiv(Bn * 32 * HW, 256), 256, 0, stream>>>(c1, cab_w2, cab_b2, c2, Bn, 8, 32, 96, 96, 0);
  // 4-5. SE
  k_se_pool<<<Bn * 32, 256, 0, stream>>>(c2, sem, HW);
  k_se_gate<<<Bn, 32, 0, stream>>>(sem, cab_aw1, cab_ab1, cab_aw2, cab_ab2, sey);
  // 6. sq1 1x1 (32->8) via WMMA GEMM
  k_gemm1x1<8, 32><<<dim3(576, Bn), 32, 0, stream>>>(sq1_w, sq1_b, xn, nullptr, zsq1, HW);
  // 7. sq2 grouped 2x2 s2
  k_sq2<<<cdiv(Bn * 16 * n2, 256), 256, 0, stream>>>(zsq1, sq2_w, sq2_b, zsq2);
  // 8. channel attention (IDX folded)
  k_chan_attn<<<dim3(16, Bn), 256, 0, stream>>>(zsq2, ca_qkv_w, ca_qkv_b, ca_t, oca);
  // 9. ca fuse
  k_fuse<<<cdiv(Bn * 16 * n2, 256), 256, 0, stream>>>(oca, ca_fus_w, ca_fus_b, zca, Bn * 16 * n2);
  // 10. spatial flash attention (NIDX folded) -- WMMA
  k_spat_attn<<<dim3(144, 16, Bn), 32, 0, stream>>>(zca, sa_qkv_w, sa_qkv_b, sa_t, osa);
  // 11. sa fuse
  k_fuse<<<cdiv(Bn * 16 * n2, 256), 256, 0, stream>>>(osa, sa_fus_w, sa_fus_b, zsa, Bn * 16 * n2);
  // 12. un1 grouped 1x1
  k_un1<<<cdiv(Bn * 32 * n2, 256), 256, 0, stream>>>(zsa, un1_w, un1_b, un1o, Bn * 32 * n2);
  // 13. pixel shuffle
  k_shuffle<<<cdiv(Bn * 8 * HW, 256), 256, 0, stream>>>(un1o, shuf, Bn * 8 * HW);
  // 14. un2 1x1 (8->32) via WMMA GEMM
  k_gemm1x1<32, 8><<<dim3(576, Bn), 32, 0, stream>>>(un2_w, un2_b, shuf, nullptr, zfin, HW);
  // 15. residual: x2 = x + z + c2*se
  k_resadd<<<cdiv(Bn * 32 * HW, 256), 256, 0, stream>>>(x, zfin, c2, sey, x2, Bn * 32 * HW);
  // 16. LN2
  k_ln<<<cdiv(Bn * HW, 256), 256, 0, stream>>>(x2, ln2_w, ln2_b, xn2, Bn * HW, 32, HW);
  // 17. da_w1 1x1 (32->32) via WMMA GEMM
  k_gemm1x1<32, 32><<<dim3(576, Bn), 32, 0, stream>>>(da_w1, da_b1, xn2, nullptr, d1, HW);
  // 18. depthwise 7x7 + gelu*sigmoid gate
  k_dwgate<<<cdiv(Bn * 32 * HW, 256), 256, 0, stream>>>(d1, da_dw_w, da_dw_b, dact, Bn * 32 * HW);
  // 19. da_pw 1x1 (32->32) + residual x2 -> final output (WMMA GEMM epilogue)
  k_gemm1x1<32, 32><<<dim3(576, Bn), 32, 0, stream>>>(da_pw, da_pb, dact, x2, out, HW);
  (void)in_sizes; (void)n_in; (void)out_size; (void)ws_size;
}